// LinearAttentionICL_22162031248049
// MI455X (gfx1250) — compile-verified
//
#include <hip/hip_runtime.h>
#include <hip/hip_bf16.h>

typedef float v2f __attribute__((ext_vector_type(2)));
typedef float v8f __attribute__((ext_vector_type(8)));

// ---------------- problem constants (reference setup is fixed) -------------
#define DDIM   512
#define HID    513
#define PLEN   8192
#define KTEST  2048
#define LDEPTH 8
#define HP     576          // HID padded to multiple of 64

#define TILE_K 16
#define KPAIRS (TILE_K / 2)
#define BM 64
#define BN 64
#define RS 160              // LDS floats per k-pair row: 64*2 + 32 pad
                            // -> lanes 0-15 (kpair q) and 16-31 (kpair q+1) hit
                            //    disjoint 32-bank halves on ds_load_b64

// ---------------- generic fp32 WMMA GEMM -----------------------------------
// C[M,N] (row-major, ldc=N) = alpha * A_op @ B_op + beta * Cin
// A_op[m][k] = A[m*rsA + k*csA], B_op[k][n] = B[k*rsB + n*csB]
// M,N multiples of 64; K multiple of 16. 128 threads = 4 waves per block.
__global__ __launch_bounds__(128) void gemm_wmma_f32(
    const float* __restrict__ A, long rsA, long csA,
    const float* __restrict__ B, long rsB, long csB,
    const float* __restrict__ Cin, float alpha, float beta,
    float* __restrict__ C, int M, int N, int K)
{
  // k-paired LDS layout: Xs2[kpair*RS + idx*2 + (k&1)]
  __shared__ float As2[KPAIRS * RS];
  __shared__ float Bs2[KPAIRS * RS];

  const int t    = threadIdx.x;       // 0..127
  const int wave = t >> 5;            // 0..3
  const int lane = t & 31;
  const int lrow = lane & 15;
  const int hi   = lane >> 4;         // 0/1 half-wave

  const int bm = blockIdx.y * BM;
  const int bn = blockIdx.x * BN;
  const int wr = (wave >> 1) * 32;    // wave row offset in block tile
  const int wc = (wave & 1)  * 32;    // wave col offset in block tile

  // ---- strength-reduced fill addressing (idx = t + 128r collapses to): ----
  // A: kp = r, mm = t>>1, kb = t&1   -> one base ptr, +2*csA per r, +16*csA per tile
  // B: kk = 2r + (t>>6), nn = t&63   -> one base ptr, +2*rsB per r, +16*rsB per tile
  const float* pA = A + (long)(bm + (t >> 1)) * rsA + (long)(t & 1) * csA;
  const float* pB = B + (long)(t >> 6) * rsB + (long)(bn + (t & 63)) * csB;
  float* sA = &As2[(t >> 1) * 2 + (t & 1)];          // + r*RS per r (invariant)
  float* sB = &Bs2[(t & 63) * 2 + (t >> 6)];         // + r*RS per r (invariant)
  const long csA2 = 2 * csA, rsB2 = 2 * rsB;
  const long stepA = (long)TILE_K * csA;
  const long stepB = (long)TILE_K * rsB;

  v8f acc[2][2] = {};                 // 2x2 fragments of 16x16

  for (int k0 = 0; k0 < K; k0 += TILE_K) {
#pragma unroll
    for (int r = 0; r < 8; ++r)
      sA[r * RS] = pA[r * csA2];
#pragma unroll
    for (int r = 0; r < 8; ++r)
      sB[r * RS] = pB[r * rsB2];
    pA += stepA;
    pB += stepB;
    // prefetch next K tile while this one is consumed (L2-resident data)
    if (k0 + TILE_K < K) {
      __builtin_prefetch(pA, 0, 1);
      __builtin_prefetch(pB, 0, 1);
    }
    __syncthreads();

#pragma unroll
    for (int ks = 0; ks < TILE_K; ks += 4) {
      const int q = (ks >> 1) + hi;   // k-pair row for this half-wave
      // fragment = one ds_load_b64: {X[..][ks+2hi], X[..][ks+2hi+1]}
      v2f a[2], b[2];
#pragma unroll
      for (int mi = 0; mi < 2; ++mi) {
        int m = wr + mi * 16 + lrow;
        a[mi] = *(const v2f*)&As2[q * RS + m * 2];
      }
#pragma unroll
      for (int ni = 0; ni < 2; ++ni) {
        int n = wc + ni * 16 + lrow;
        b[ni] = *(const v2f*)&Bs2[q * RS + n * 2];
      }
#pragma unroll
      for (int mi = 0; mi < 2; ++mi)
#pragma unroll
        for (int ni = 0; ni < 2; ++ni)
          acc[mi][ni] = __builtin_amdgcn_wmma_f32_16x16x4_f32(
              false, a[mi], false, b[ni], (short)0, acc[mi][ni], false, false);
    }
    __syncthreads();
  }

  // epilogue: D element [m][n]: vgpr v = (m%16)&7, lane = 16*((m%16)>=8)+n
#pragma unroll
  for (int mi = 0; mi < 2; ++mi) {
#pragma unroll
    for (int ni = 0; ni < 2; ++ni) {
#pragma unroll
      for (int v = 0; v < 8; ++v) {
        int m = bm + wr + mi * 16 + v + 8 * hi;
        int n = bn + wc + ni * 16 + lrow;
        long off = (long)m * N + n;
        float val = alpha * acc[mi][ni][v];
        if (beta != 0.0f) val += beta * Cin[off];
        C[off] = val;
      }
    }
  }
}

// ---------------- helpers ---------------------------------------------------
__global__ void zero_f32(float* p, long n) {
  long i = (long)blockIdx.x * blockDim.x + threadIdx.x;
  long stride = (long)gridDim.x * blockDim.x;
  for (; i < n; i += stride) p[i] = 0.0f;
}

__global__ void copy_f32(float* dst, const float* src, long n) {
  long i = (long)blockIdx.x * blockDim.x + threadIdx.x;
  long stride = (long)gridDim.x * blockDim.x;
  for (; i < n; i += stride) dst[i] = src[i];
}

// pad 513x513 row-major -> 576x576 row-major (rest already zeroed)
__global__ void pad_hid_mat(float* dst, const float* src) {
  long i = (long)blockIdx.x * blockDim.x + threadIdx.x;
  long n = (long)HID * HID;
  long stride = (long)gridDim.x * blockDim.x;
  for (; i < n; i += stride) {
    long r = i / HID, c = i % HID;
    dst[r * HP + c] = src[i];
  }
}

// H[i][j] += wy[i] * y[j]   (H is HP x PLEN; wy zero-padded)
__global__ void add_rank1(float* H, const float* wy, const float* y) {
  int col = blockIdx.x * blockDim.x + threadIdx.x;
  int row = blockIdx.y;
  H[(long)row * PLEN + col] += wy[row] * y[col];
}

// out[j] = sum_i wo[i] * Hte[i][j]   (Hte is HP x KTEST; wo zero-padded)
__global__ void reduce_out(float* out, const float* Hte, const float* wo) {
  int j = blockIdx.x * blockDim.x + threadIdx.x;
  float s = 0.0f;
  for (int i = 0; i < HP; ++i) s += wo[i] * Hte[(long)i * KTEST + j];
  out[j] = s;
}

// ---------------- launch -----------------------------------------------------
static inline void launch_gemm(hipStream_t st,
                               const float* A, long rsA, long csA,
                               const float* B, long rsB, long csB,
                               const float* Cin, float alpha, float beta,
                               float* C, int M, int N, int K) {
  dim3 grid(N / BN, M / BM);
  gemm_wmma_f32<<<grid, 128, 0, st>>>(A, rsA, csA, B, rsB, csB, Cin,
                                      alpha, beta, C, M, N, K);
}

extern "C" void kernel_launch(void* const* d_in, const int* in_sizes, int n_in,
                              void* d_out, int out_size, void* d_ws, size_t ws_size,
                              hipStream_t stream) {
  const float* X     = (const float*)d_in[0];   // (P, D)
  const float* y     = (const float*)d_in[1];   // (P,)
  const float* Xs    = (const float*)d_in[2];   // (KTEST, D)
  const float* W_x   = (const float*)d_in[3];   // (HID, D)
  const float* w_y   = (const float*)d_in[4];   // (HID,)
  const float* w_o   = (const float*)d_in[5];   // (HID,)
  const float* W_k   = (const float*)d_in[6];   // (HID, HID)
  const float* W_q   = (const float*)d_in[7];   // (HID, HID)
  const float* W_v   = (const float*)d_in[8];   // (HID, HID)
  float* out = (float*)d_out;                   // (KTEST,)

  // ---- workspace layout (floats), all zero-padded shapes -------------------
  float* ws = (float*)d_ws;
  long off = 0;
  float* Wxp  = ws + off; off += (long)HP * DDIM;      // 576x512
  float* Wkp  = ws + off; off += (long)HP * HP;        // 576x576
  float* Wqp  = ws + off; off += (long)HP * HP;
  float* Wvp  = ws + off; off += (long)HP * HP;
  float* wyp  = ws + off; off += HP;
  float* wop  = ws + off; off += HP;
  float* Mb   = ws + off; off += (long)HP * HP;        // M = scale*V K^T
  float* Ab   = ws + off; off += (long)HP * HP;        // A = M Wq
  float* Htr0 = ws + off; off += (long)HP * PLEN;
  float* Htr1 = ws + off; off += (long)HP * PLEN;
  float* Ktr  = ws + off; off += (long)HP * PLEN;
  float* Vtr  = ws + off; off += (long)HP * PLEN;
  float* Hte0 = ws + off; off += (long)HP * KTEST;
  float* Hte1 = ws + off; off += (long)HP * KTEST;
  const long total = off;                              // ~23.2M floats (~93 MB)

  // zero workspace (padding must be zero every call; harness poisons d_ws)
  zero_f32<<<2048, 256, 0, stream>>>(ws, total);

  // pack inputs into padded buffers
  copy_f32<<<512, 256, 0, stream>>>(Wxp, W_x, (long)HID * DDIM); // rows same ld
  pad_hid_mat<<<512, 256, 0, stream>>>(Wkp, W_k);
  pad_hid_mat<<<512, 256, 0, stream>>>(Wqp, W_q);
  pad_hid_mat<<<512, 256, 0, stream>>>(Wvp, W_v);
  copy_f32<<<4, 256, 0, stream>>>(wyp, w_y, HID);
  copy_f32<<<4, 256, 0, stream>>>(wop, w_o, HID);

  // H_train = W_x @ X^T + w_y ⊗ y      (X^T element [k][j] = X[j*D + k])
  launch_gemm(stream, Wxp, DDIM, 1, X, 1, DDIM, Htr0, 1.0f, 0.0f,
              Htr0, HP, PLEN, DDIM);
  add_rank1<<<dim3(PLEN / 256, HP), 256, 0, stream>>>(Htr0, wyp, y);

  // H_test = W_x @ X_star^T
  launch_gemm(stream, Wxp, DDIM, 1, Xs, 1, DDIM, Hte0, 1.0f, 0.0f,
              Hte0, HP, KTEST, DDIM);

  const float scale = 1.0f / ((float)LDEPTH * (float)PLEN);
  float* HtrBuf[2] = {Htr0, Htr1};
  float* HteBuf[2] = {Hte0, Hte1};

  for (int l = 0; l < LDEPTH; ++l) {
    float* Hc  = HtrBuf[l & 1];
    float* Hn  = HtrBuf[(l + 1) & 1];
    float* Tc  = HteBuf[l & 1];
    float* Tn  = HteBuf[(l + 1) & 1];

    // K_tr = W_k @ H,  V_tr = W_v @ H        (HP x PLEN)
    launch_gemm(stream, Wkp, HP, 1, Hc, PLEN, 1, Ktr, 1.0f, 0.0f,
                Ktr, HP, PLEN, HP);
    launch_gemm(stream, Wvp, HP, 1, Hc, PLEN, 1, Vtr, 1.0f, 0.0f,
                Vtr, HP, PLEN, HP);
    // M = scale * V_tr @ K_tr^T             (HP x HP, reduce over P)
    launch_gemm(stream, Vtr, PLEN, 1, Ktr, 1, PLEN, Mb, scale, 0.0f,
                Mb, HP, HP, PLEN);
    // A = M @ W_q                            (HP x HP)
    launch_gemm(stream, Mb, HP, 1, Wqp, HP, 1, Ab, 1.0f, 0.0f,
                Ab, HP, HP, HP);
    // H_train' = H_train - A @ H_train
    launch_gemm(stream, Ab, HP, 1, Hc, PLEN, 1, Hc, -1.0f, 1.0f,
                Hn, HP, PLEN, HP);
    // H_test'  = H_test  + A @ H_test
    launch_gemm(stream, Ab, HP, 1, Tc, KTEST, 1, Tc, 1.0f, 1.0f,
                Tn, HP, KTEST, HP);
  }

  // out = w_o @ H_test   (final H_test in buffer index LDEPTH&1 == 0)
  reduce_out<<<KTEST / 256, 256, 0, stream>>>(out, HteBuf[LDEPTH & 1], wop);
}